// ProtoPNetClassifier_24395414241700
// MI455X (gfx1250) — compile-verified
//
#include <hip/hip_runtime.h>
#include <hip/hip_bf16.h>

#define NUM_CLASSES 200
#define NUM_FEATURES 128
#define NUM_PROTOTYPES 2000
#define NUM_IMAGES 64
#define HW 196                 // 14*14
#define EPS_VAL 1e-4f
#define PTILES (NUM_PROTOTYPES / 16)   // 125
#define PXTILES 13                     // ceil(196/16)
#define FLT_BIG 3.402823466e38f

typedef __attribute__((ext_vector_type(2))) float v2f;
typedef __attribute__((ext_vector_type(8))) float v8f;

// One wave per (image n, tile of 16 prototypes).
// Computes min over all 196 pixels of relu(||x||^2 + ||p||^2 - 2 x.p) using
// V_WMMA_F32_16X16X4_F32 for the x.p GEMM (K=128 in 32 steps of 4).
__global__ __launch_bounds__(32)
void proto_min_dist_kernel(const float* __restrict__ features,   // (64,128,14,14)
                           const float* __restrict__ protos,     // (2000,128)
                           float* __restrict__ min_dist_out) {   // (64,2000)
    const int pt   = blockIdx.x;        // prototype tile 0..124
    const int n    = blockIdx.y;        // image 0..63
    const int lane = threadIdx.x;       // 0..31
    const int lo   = lane & 15;
    const int hi   = lane >> 4;         // 0 or 1

    // ---- A panel: 16 protos x 128 K in f32-WMMA A layout ----
    // lane (hi,lo): M = lo ; VGPR pair s holds K = 4s+2hi, 4s+2hi+1
    const int p = pt * 16 + lo;
    const float* prow = protos + (size_t)p * NUM_FEATURES + 2 * hi;
    v2f a[32];
    float psq_part = 0.0f;
#pragma unroll
    for (int s = 0; s < 32; ++s) {
        a[s] = *(const v2f*)(prow + 4 * s);
        psq_part += a[s].x * a[s].x + a[s].y * a[s].y;
    }
    // each lane holds half the K range of proto `lo`; merge with partner lane
    float psq = psq_part + __shfl_xor(psq_part, 16, 32);
    // this lane-half accumulates rows M = 8*hi .. 8*hi+7 -> fetch their ||p||^2
    float psq_m[8];
#pragma unroll
    for (int r = 0; r < 8; ++r) psq_m[r] = __shfl(psq, r + 8 * hi, 32);

    const float* fbase = features + (size_t)n * (NUM_FEATURES * HW);
    float vmin[8];
#pragma unroll
    for (int r = 0; r < 8; ++r) vmin[r] = FLT_BIG;

    for (int t = 0; t < PXTILES; ++t) {
        const int px    = t * 16 + lo;
        const bool valid = (px < HW);
        const int pxc   = valid ? px : (HW - 1);       // clamp: no OOB, masked below
        const float* fcol = fbase + pxc + (size_t)(2 * hi) * HW;

        v8f acc = {};
        float bsq = 0.0f;
#pragma unroll
        for (int s = 0; s < 32; ++s) {
            // B layout (4x16 K x N): VGPR0 = K=4s+2hi, VGPR1 = K=4s+2hi+1; lane col = lo
            v2f b;
            b.x = fcol[(size_t)(4 * s)     * HW];
            b.y = fcol[(size_t)(4 * s + 1) * HW];
            bsq += b.x * b.x + b.y * b.y;
            // D = A(16x4) * B(4x16) + C   (args: neg_a, A, neg_b, B, c_mod, C, reuse_a, reuse_b)
            acc = __builtin_amdgcn_wmma_f32_16x16x4_f32(false, a[s], false, b,
                                                        (short)0, acc, false, false);
        }
        // ||x||^2 for this lane's pixel: merge complementary K halves
        float xsq = bsq + __shfl_xor(bsq, 16, 32);
#pragma unroll
        for (int r = 0; r < 8; ++r) {
            float d = fmaxf(xsq + psq_m[r] - 2.0f * acc[r], 0.0f);  // relu
            d = valid ? d : FLT_BIG;                                // mask padded pixels
            vmin[r] = fminf(vmin[r], d);
        }
    }

    // min over the 16 pixel-columns held by this lane-half
#pragma unroll
    for (int m = 1; m <= 8; m <<= 1) {
#pragma unroll
        for (int r = 0; r < 8; ++r)
            vmin[r] = fminf(vmin[r], __shfl_xor(vmin[r], m, 32));
    }
    if (lo == 0) {  // lanes 0 and 16 own protos pt*16+{0..7} / +{8..15}
        float* outrow = min_dist_out + (size_t)n * NUM_PROTOTYPES + pt * 16 + 8 * hi;
#pragma unroll
        for (int r = 0; r < 8; ++r) outrow[r] = vmin[r];
    }
}

// sim_max = log((min_dist+1)/(min_dist+eps)) (sim is monotone-decreasing in dist),
// then logits[n,c] = sum_p sim[n,p] * W[c,p]. One block per image.
__global__ __launch_bounds__(256)
void proto_logits_kernel(const float* __restrict__ min_dist,   // (64,2000)
                         const float* __restrict__ W,          // (200,2000)
                         float* __restrict__ logits) {         // (64,200)
    __shared__ float sim[NUM_PROTOTYPES];
    const int n = blockIdx.x;
    const float* mdrow = min_dist + (size_t)n * NUM_PROTOTYPES;
    for (int q = threadIdx.x; q < NUM_PROTOTYPES; q += blockDim.x) {
        float md = mdrow[q];
        sim[q] = logf((md + 1.0f) / (md + EPS_VAL));
    }
    __syncthreads();
    for (int c = threadIdx.x; c < NUM_CLASSES; c += blockDim.x) {
        const float* wrow = W + (size_t)c * NUM_PROTOTYPES;
        float s = 0.0f;
        for (int q = 0; q < NUM_PROTOTYPES; ++q) s = fmaf(sim[q], wrow[q], s);
        logits[(size_t)n * NUM_CLASSES + c] = s;
    }
}

extern "C" void kernel_launch(void* const* d_in, const int* in_sizes, int n_in,
                              void* d_out, int out_size, void* d_ws, size_t ws_size,
                              hipStream_t stream) {
    (void)in_sizes; (void)n_in; (void)d_ws; (void)ws_size; (void)out_size;
    const float* features = (const float*)d_in[0];   // (64,128,14,14)
    const float* protos   = (const float*)d_in[1];   // (2000,128,1,1)
    const float* lastW    = (const float*)d_in[2];   // (200,2000)

    float* logits   = (float*)d_out;                           // first 64*200
    float* min_dist = (float*)d_out + NUM_IMAGES * NUM_CLASSES; // next 64*2000

    dim3 grid1(PTILES, NUM_IMAGES);
    proto_min_dist_kernel<<<grid1, 32, 0, stream>>>(features, protos, min_dist);
    proto_logits_kernel<<<NUM_IMAGES, 256, 0, stream>>>(min_dist, lastW, logits);
}